// MultiScaleRoIAlign_37950331028056
// MI455X (gfx1250) — compile-verified
//
#include <hip/hip_runtime.h>
#include <stdint.h>

// ---------------------------------------------------------------------------
// MultiScaleRoIAlign for MI455X (gfx1250).
//
// Strategy: gather-bound op. Feature maps (~177MB) fit in 192MB L2. Per RoI,
// DMA the bounding region of all bilinear taps into LDS with the Tensor Data
// Mover (tensor_load_to_lds, 3D tile: x,y,channel-chunk), double-buffered on
// TENSORcnt, then do the 7x7x(2x2) bilinear pooling out of LDS with VALU.
// WMMA is deliberately NOT used: total math is ~0.4 GFLOP and a dense-GEMM
// formulation inflates FLOPs ~25x for a memory-bound problem.
// ---------------------------------------------------------------------------

#define ROUT   7
#define NBIN   49          // 7*7
#define NGRID  14          // 7 bins * 2 samples per axis
#define CTOT   256         // channels
#define CHUNK  8           // channels staged per TDM tile
#define NCHUNK (CTOT / CHUNK)
#define CAP    960         // max region elements/channel for LDS path
                           // (FPN level assignment bounds region to <=131x7)

typedef uint32_t v4u __attribute__((ext_vector_type(4)));
typedef int32_t  v4i __attribute__((ext_vector_type(4)));
typedef int32_t  v8i __attribute__((ext_vector_type(8)));

__device__ __forceinline__ uint32_t rfl(uint32_t x) {
    return (uint32_t)__builtin_amdgcn_readfirstlane((int)x);
}

// Matches reference _interp_1d; validity folded into caller's weights.
__device__ __forceinline__ void interp1d(float c, int size,
                                         int& lo, int& hi, float& fr, float& valid) {
    valid = (c >= -1.0f && c <= (float)size) ? 1.0f : 0.0f;
    float cc   = fmaxf(c, 0.0f);
    float lo_f = floorf(cc);
    bool  edge = lo_f >= (float)(size - 1);
    lo = edge ? (size - 1) : (int)lo_f;
    hi = edge ? (size - 1) : ((int)lo_f + 1);
    fr = edge ? 0.0f : (cc - lo_f);
}

// Issue one TDM 3D tile load: [CHUNK channels][hreg rows][wreg cols] -> LDS.
// D# bitfields per CDNA5 ISA 8.3-8.6. 6-arg clang-23 builtin form:
// (uint32x4 g0, int32x8 g1, int32x4 g2, int32x4 g3, int32x8 pad, i32 cpol).
__device__ __forceinline__ void tdm_issue(const float* gptr, uint32_t lds_off,
                                          int wreg, int hreg, int W, uint64_t plane) {
    uint64_t ga = (uint64_t)(uintptr_t)gptr;
    uint64_t s0 = (uint64_t)(uint32_t)W;   // dim0 stride: row (elements)
    uint64_t s1 = plane;                   // dim1 stride: channel (H*W elements)
    uint32_t d0 = (uint32_t)wreg;          // tensor_dim0 == tile_dim0
    uint32_t d1 = (uint32_t)hreg;          // tensor_dim1 == tile_dim1

    v4u g0;
    g0.x = rfl(1u);                                      // count=1, user mode
    g0.y = rfl(lds_off);                                 // lds_addr (bytes)
    g0.z = rfl((uint32_t)ga);                            // global_addr[31:0]
    g0.w = rfl(((uint32_t)(ga >> 32) & 0x1FFFFFFu)       // global_addr[56:32]
               | (2u << 30));                            // type=2 ("image")

    v8i g1;
    g1[0] = (int)rfl(2u << 16);                          // data_size=2 (4B), mask=0
    g1[1] = (int)rfl((d0 & 0xFFFFu) << 16);              // tensor_dim0[15:0]
    g1[2] = (int)rfl((d0 >> 16) | ((d1 & 0xFFFFu) << 16));
    g1[3] = (int)rfl((d1 >> 16) | ((uint32_t)wreg << 16));   // tile_dim0
    g1[4] = (int)rfl((uint32_t)hreg | ((uint32_t)CHUNK << 16)); // tile_dim1, tile_dim2
    g1[5] = (int)rfl((uint32_t)s0);                      // dim0_stride[31:0]
    g1[6] = (int)rfl(((uint32_t)(s0 >> 32) & 0xFFFFu)
                     | (((uint32_t)s1 & 0xFFFFu) << 16));// dim0_str[47:32]|dim1_str[15:0]
    g1[7] = (int)rfl((uint32_t)(s1 >> 16));              // dim1_stride[47:16]

    v4i g2;
    g2[0] = (int)rfl((uint32_t)CHUNK);                   // tensor_dim2
    g2[1] = 0; g2[2] = 0; g2[3] = 0;

    v4i g3; g3[0] = 0; g3[1] = 0; g3[2] = 0; g3[3] = 0;

    v8i gz;                                              // reserved/unused group
    gz[0] = 0; gz[1] = 0; gz[2] = 0; gz[3] = 0;
    gz[4] = 0; gz[5] = 0; gz[6] = 0; gz[7] = 0;

    __builtin_amdgcn_tensor_load_to_lds(g0, g1, g2, g3, gz, /*cpol*/0);
}

__global__ __launch_bounds__(256)
void roi_align_kernel(const float* __restrict__ f0, const float* __restrict__ f1,
                      const float* __restrict__ f2, const float* __restrict__ f3,
                      const float* __restrict__ bboxes, float* __restrict__ out,
                      int N, int R) {
    __shared__ float sbuf[2][CHUNK * CAP];               // 61440 B, double-buffered
    __shared__ int   s_xl[NGRID], s_xh[NGRID], s_yl[NGRID], s_yh[NGRID];
    __shared__ float s_wx0[NGRID], s_wx1[NGRID], s_wy0[NGRID], s_wy1[NGRID];

    const int roi = blockIdx.x;
    const int t   = threadIdx.x;
    if (roi >= N) return;
    const int b = roi / R;

    const float4 bb = ((const float4*)bboxes)[roi];
    const float bx1 = bb.x, by1 = bb.y, bx2 = bb.z, by2 = bb.w;
    const float area = (by2 - by1) * (bx2 - bx1);
    const int level = (area < 384.0f * 384.0f) + (area < 192.0f * 192.0f)
                    + (area < 96.0f * 96.0f);

    const float* feat; int H; float scale;
    if      (level == 0) { feat = f0; H = 32;  scale = 1.0f / 32.0f; }
    else if (level == 1) { feat = f1; H = 64;  scale = 1.0f / 16.0f; }
    else if (level == 2) { feat = f2; H = 128; scale = 1.0f / 8.0f;  }
    else                 { feat = f3; H = 256; scale = 1.0f / 4.0f;  }
    const int W = H;

    const float x1 = bx1 * scale, y1 = by1 * scale;
    const float x2 = bx2 * scale, y2 = by2 * scale;
    const float roi_w = fmaxf(x2 - x1, 1.0f), roi_h = fmaxf(y2 - y1, 1.0f);
    const float bin_w = roi_w / (float)ROUT, bin_h = roi_h / (float)ROUT;

    // Per-axis interpolation tables: grid_i = (i>>1) + ((i&1)+0.5)/2
    if (t < NGRID) {
        const int i = t;
        const float g  = (float)(i >> 1) + ((float)(i & 1) + 0.5f) * 0.5f;
        const float xs = x1 + g * bin_w;
        int lo, hi; float fr, vv;
        interp1d(xs, W, lo, hi, fr, vv);
        s_xl[i] = lo; s_xh[i] = hi;
        s_wx0[i] = (1.0f - fr) * vv; s_wx1[i] = fr * vv;
    } else if (t >= 32 && t < 32 + NGRID) {
        const int i = t - 32;
        const float g  = (float)(i >> 1) + ((float)(i & 1) + 0.5f) * 0.5f;
        const float ys = y1 + g * bin_h;
        int lo, hi; float fr, vv;
        interp1d(ys, H, lo, hi, fr, vv);
        s_yl[i] = lo; s_yh[i] = hi;
        s_wy0[i] = (1.0f - fr) * vv; s_wy1[i] = fr * vv;
    }
    __syncthreads();

    // Bounding region of all taps (grid coords are monotonic per axis).
    const int x0 = s_xl[0], xe = s_xh[NGRID - 1];
    const int y0 = s_yl[0], ye = s_yh[NGRID - 1];
    const int wreg = xe - x0 + 1, hreg = ye - y0 + 1;
    const int reg  = wreg * hreg;

    const uint64_t plane  = (uint64_t)H * (uint64_t)W;
    const uint64_t base_b = (uint64_t)b * CTOT * plane;
    float* const out_roi  = out + (uint64_t)roi * (uint64_t)(CTOT * NBIN);

    if (reg <= CAP) {
        // ---- TDM-staged path: double-buffered channel chunks ----
        const float* g_region = feat + base_b + (uint64_t)(y0 * W + x0);
        if (t < 32) {
            tdm_issue(g_region, (uint32_t)(uintptr_t)(void*)&sbuf[0][0],
                      wreg, hreg, W, plane);
        }
        for (int k = 0; k < NCHUNK; ++k) {
            if (t < 32) {
                if (k + 1 < NCHUNK) {
                    tdm_issue(g_region + (uint64_t)(k + 1) * CHUNK * plane,
                              (uint32_t)(uintptr_t)(void*)&sbuf[(k + 1) & 1][0],
                              wreg, hreg, W, plane);
                    __builtin_amdgcn_s_wait_tensorcnt(1); // chunk k complete
                } else {
                    __builtin_amdgcn_s_wait_tensorcnt(0);
                }
            }
            __syncthreads();
            const float* lbuf = &sbuf[k & 1][0];

            for (int p = t; p < CHUNK * NBIN; p += 256) {
                const int c   = p & (CHUNK - 1);
                const int bin = p >> 3;
                const int oy = bin / ROUT, ox = bin - oy * ROUT;
                const float* cb = lbuf + (size_t)c * (size_t)reg;
                float acc = 0.0f;
                #pragma unroll
                for (int sy = 0; sy < 2; ++sy) {
                    const int gy  = oy * 2 + sy;
                    const int ryl = s_yl[gy] - y0, ryh = s_yh[gy] - y0;
                    const float wy0 = s_wy0[gy], wy1 = s_wy1[gy];
                    #pragma unroll
                    for (int sx = 0; sx < 2; ++sx) {
                        const int gx  = ox * 2 + sx;
                        const int rxl = s_xl[gx] - x0, rxh = s_xh[gx] - x0;
                        const float wx0 = s_wx0[gx], wx1 = s_wx1[gx];
                        const float v00 = cb[ryl * wreg + rxl];
                        const float v01 = cb[ryl * wreg + rxh];
                        const float v10 = cb[ryh * wreg + rxl];
                        const float v11 = cb[ryh * wreg + rxh];
                        acc += wy0 * (wx0 * v00 + wx1 * v01)
                             + wy1 * (wx0 * v10 + wx1 * v11);
                    }
                }
                out_roi[(size_t)(k * CHUNK + c) * NBIN + bin] = 0.25f * acc;
            }
            __syncthreads();
        }
    } else {
        // ---- Fallback: direct global gather (region too big for LDS) ----
        for (int k = 0; k < NCHUNK; ++k) {
            for (int p = t; p < CHUNK * NBIN; p += 256) {
                const int c   = p & (CHUNK - 1);
                const int bin = p >> 3;
                const int oy = bin / ROUT, ox = bin - oy * ROUT;
                const float* cb = feat + base_b + (uint64_t)(k * CHUNK + c) * plane;
                float acc = 0.0f;
                #pragma unroll
                for (int sy = 0; sy < 2; ++sy) {
                    const int gy = oy * 2 + sy;
                    const int yl = s_yl[gy], yh = s_yh[gy];
                    const float wy0 = s_wy0[gy], wy1 = s_wy1[gy];
                    #pragma unroll
                    for (int sx = 0; sx < 2; ++sx) {
                        const int gx = ox * 2 + sx;
                        const int xl = s_xl[gx], xh = s_xh[gx];
                        const float wx0 = s_wx0[gx], wx1 = s_wx1[gx];
                        const float v00 = cb[yl * W + xl];
                        const float v01 = cb[yl * W + xh];
                        const float v10 = cb[yh * W + xl];
                        const float v11 = cb[yh * W + xh];
                        acc += wy0 * (wx0 * v00 + wx1 * v01)
                             + wy1 * (wx0 * v10 + wx1 * v11);
                    }
                }
                out_roi[(size_t)(k * CHUNK + c) * NBIN + bin] = 0.25f * acc;
            }
        }
    }
}

extern "C" void kernel_launch(void* const* d_in, const int* in_sizes, int n_in,
                              void* d_out, int out_size, void* d_ws, size_t ws_size,
                              hipStream_t stream) {
    const float* f0 = (const float*)d_in[0];
    const float* f1 = (const float*)d_in[1];
    const float* f2 = (const float*)d_in[2];
    const float* f3 = (const float*)d_in[3];
    const float* bb = (const float*)d_in[4];
    float* out = (float*)d_out;

    const int B = in_sizes[0] / (CTOT * 32 * 32);   // feat0 is (B,256,32,32)
    const int N = in_sizes[4] / 4;                  // bboxes is (B,R,4)
    const int R = N / (B > 0 ? B : 1);

    dim3 grid((unsigned)N), block(256);
    hipLaunchKernelGGL(roi_align_kernel, grid, block, 0, stream,
                       f0, f1, f2, f3, bb, out, N, R);
}